// CapsulePoolingGraphNetwork_32392643346849
// MI455X (gfx1250) — compile-verified
//
#include <hip/hip_runtime.h>
#include <math.h>

// ---------------- problem constants (match reference) ----------------
constexpr int NN = 50000;     // nodes
constexpr int EE = 640000;    // edges
constexpr int FH = 128;       // F_IN == HIDDEN
constexpr int NG = 256;       // graphs
constexpr int NC = 10;        // classes
constexpr int NL = 3;         // layers
constexpr float EPS_BN = 1e-5f;
constexpr float EPS_SQ = 1e-8f;

typedef __attribute__((ext_vector_type(2))) float v2f;
typedef __attribute__((ext_vector_type(8))) float v8f;

// ---------------- init everything we accumulate into ----------------
__global__ void setup_init(float* rep, float* wsum, float* msum, float* mmax,
                           float* deg, float* counts, float* sums, float* sumsq) {
    int i = blockIdx.x * blockDim.x + threadIdx.x;
    if (i < NG * 3 * FH) rep[i] = 0.f;
    if (i < NG * FH) { wsum[i] = 0.f; msum[i] = 0.f; mmax[i] = -INFINITY; }
    if (i < NN) deg[i] = 1.0f;                 // self-loop weight 1
    if (i < NG) counts[i] = 0.f;
    if (i < FH) { sums[i] = 0.f; sumsq[i] = 0.f; }
}

// ---------------- BatchNorm statistics (population mean/var) ----------------
__global__ __launch_bounds__(128) void bn_stats(const float* __restrict__ x,
                                                float* sums, float* sumsq) {
    int f = threadIdx.x;
    float s = 0.f, s2 = 0.f;
    for (int n = blockIdx.x; n < NN; n += gridDim.x) {
        float v = x[(size_t)n * FH + f];
        s += v; s2 += v * v;
    }
    atomicAdd(&sums[f], s);
    atomicAdd(&sumsq[f], s2);
}

__global__ void bn_apply(const float* __restrict__ x, const float* __restrict__ sums,
                         const float* __restrict__ sumsq, const float* __restrict__ gamma,
                         const float* __restrict__ beta, float* __restrict__ xo) {
    int i = blockIdx.x * blockDim.x + threadIdx.x;
    if (i >= NN * FH) return;
    int f = i & (FH - 1);
    float mu = sums[f] * (1.0f / NN);
    float var = sumsq[f] * (1.0f / NN) - mu * mu;
    xo[i] = (x[i] - mu) * rsqrtf(var + EPS_BN) * gamma[f] + beta[f];
}

// ---------------- degree (self loops folded into init=1.0) ----------------
__global__ void deg_edges(const int* __restrict__ ei, float* deg) {
    int e = blockIdx.x * blockDim.x + threadIdx.x;
    if (e >= EE) return;
    int s = ei[e], d = ei[EE + e];
    if (s != d) atomicAdd(&deg[d], 1.0f);
}

__global__ void deg_to_dinv(float* deg) {
    int i = blockIdx.x * blockDim.x + threadIdx.x;
    if (i < NN) deg[i] = rsqrtf(deg[i]);
}

__global__ void count_nodes(const int* __restrict__ batch, float* counts) {
    int n = blockIdx.x * blockDim.x + threadIdx.x;
    if (n < NN) atomicAdd(&counts[batch[n]], 1.0f);
}

// ---------------- h = x @ W  via V_WMMA_F32_16X16X4_F32 ----------------
// Wave computes a 16x128 tile of H. W (128x128) staged transposed in LDS with
// stride 130 floats (bank-conflict pad, keeps 8B alignment for ds_load_b64).
// A fragment (16x4 fp32): lane m holds (M=m, K=2*kh+v), kh=lane>>4.
// B fragment (4x16 fp32): lane m holds (K=2*kh+v, N=m).
// C/D (16x16 fp32): VGPR j -> row j+8*kh, col m.
__global__ __launch_bounds__(256) void gemm_node_wmma(const float* __restrict__ X,
                                                      const float* __restrict__ W,
                                                      float* __restrict__ H) {
    __shared__ float Wt[FH * 130];
    int tid = threadIdx.x;
    for (int i = tid; i < FH * FH; i += 256) {
        int k = i >> 7, n = i & (FH - 1);
        Wt[n * 130 + k] = W[i];               // coalesced read, transposed store
    }
    __syncthreads();

    int wave = tid >> 5, lane = tid & 31;
    int m = lane & 15, kh = lane >> 4;
    int row0 = blockIdx.x * 128 + wave * 16;
    int grow = row0 + m;
    int crow = grow < NN ? grow : NN - 1;     // clamp (stores are guarded)
    const float* xrow = X + (size_t)crow * FH;

    v8f acc[8] = {};
#pragma unroll 4
    for (int k0 = 0; k0 < FH; k0 += 4) {
        int kk = k0 + 2 * kh;
        v2f a = *(const v2f*)(xrow + kk);      // 8B aligned (kk even)
#pragma unroll
        for (int nt = 0; nt < 8; ++nt) {
            v2f b = *(const v2f*)(&Wt[(nt * 16 + m) * 130 + kk]);
            acc[nt] = __builtin_amdgcn_wmma_f32_16x16x4_f32(
                false, a, false, b, (short)0, acc[nt], false, false);
        }
    }

#pragma unroll
    for (int j = 0; j < 8; ++j) {
        int r = row0 + j + 8 * kh;
        if (r < NN) {
#pragma unroll
            for (int nt = 0; nt < 8; ++nt)
                H[(size_t)r * FH + nt * 16 + m] = acc[nt][j];
        }
    }
}

// ---------------- agg = bias + dinv^2 * h (self loop) ----------------
__global__ void agg_init(const float* __restrict__ H, const float* __restrict__ dinv,
                         const float* __restrict__ bias, float* __restrict__ AGG) {
    int i = blockIdx.x * blockDim.x + threadIdx.x;
    if (i >= NN * FH) return;
    int n = i >> 7, f = i & (FH - 1);
    float di = dinv[n];
    AGG[i] = bias[f] + di * di * H[i];
}

// ---------------- edge scatter: one wave per edge, float4 per lane ----------------
__global__ __launch_bounds__(256) void edge_scatter(const int* __restrict__ ei,
                                                    const float* __restrict__ dinv,
                                                    const float* __restrict__ H,
                                                    float* __restrict__ AGG) {
    int e = blockIdx.x * 8 + (threadIdx.x >> 5);
    if (e >= EE) return;
    int s = ei[e], d = ei[EE + e];
    if (s == d) return;                        // removed self loops (weight 0)
    float nrm = dinv[s] * dinv[d];
    int lane = threadIdx.x & 31;
    float4 v = ((const float4*)(H + (size_t)s * FH))[lane];
    float* o = AGG + (size_t)d * FH + lane * 4;
    atomicAdd(o + 0, nrm * v.x);
    atomicAdd(o + 1, nrm * v.y);
    atomicAdd(o + 2, nrm * v.z);
    atomicAdd(o + 3, nrm * v.w);
}

// ---------------- float atomic max via monotone int trick ----------------
__device__ __forceinline__ void atomicMaxF(float* addr, float val) {
    if (val >= 0.f) atomicMax((int*)addr, __float_as_int(val));
    else            atomicMin((unsigned int*)addr, __float_as_uint(val));
}

// ---------------- squash + attention readout (one block per node) ----------------
__global__ __launch_bounds__(128) void squash_readout(const float* __restrict__ AGG,
                                                      const int* __restrict__ batch,
                                                      const float* __restrict__ watt,
                                                      float* __restrict__ Xout,
                                                      float* wsum, float* msum, float* mmax) {
    __shared__ float red[4], red2[4];
    int n = blockIdx.x, f = threadIdx.x;
    float v = AGG[(size_t)n * FH + f];

    float p = v * v;
#pragma unroll
    for (int off = 16; off; off >>= 1) p += __shfl_down(p, off, 32);
    if ((f & 31) == 0) red[f >> 5] = p;
    __syncthreads();
    float n2 = red[0] + red[1] + red[2] + red[3];
    float scale = (n2 / (1.f + n2)) * rsqrtf(n2 + EPS_SQ);
    float xo = v * scale;
    Xout[(size_t)n * FH + f] = xo;

    float q = xo * watt[f];
#pragma unroll
    for (int off = 16; off; off >>= 1) q += __shfl_down(q, off, 32);
    if ((f & 31) == 0) red2[f >> 5] = q;
    __syncthreads();
    float att = red2[0] + red2[1] + red2[2] + red2[3];

    int g = batch[n];
    atomicAdd(&wsum[g * FH + f], att * xo);
    atomicAdd(&msum[g * FH + f], xo);
    atomicMaxF(&mmax[g * FH + f], xo);
}

// ---------------- fold per-layer readout into graph_rep, reset accumulators ----------------
__global__ void layer_accum(float* rep, float* wsum, float* msum, float* mmax,
                            const float* __restrict__ counts) {
    int i = blockIdx.x * blockDim.x + threadIdx.x;
    if (i >= NG * FH) return;
    int g = i >> 7, f = i & (FH - 1);
    float c = counts[g]; c = c < 1.f ? 1.f : c;
    float w = wsum[i], m = msum[i], x = mmax[i];
    if (!(x > -INFINITY)) x = 0.f;             // empty-graph -inf -> 0
    rep[g * 3 * FH + f]            += w;
    rep[g * 3 * FH + FH + f]       += m / c;
    rep[g * 3 * FH + 2 * FH + f]   += x;
    wsum[i] = 0.f; msum[i] = 0.f; mmax[i] = -INFINITY;
}

// ---------------- MLP head + log_softmax (one block per graph) ----------------
__global__ __launch_bounds__(128) void mlp_head(const float* __restrict__ rep,
                                                const float* __restrict__ W1,
                                                const float* __restrict__ b1,
                                                const float* __restrict__ W2,
                                                const float* __restrict__ b2,
                                                float* __restrict__ out) {
    __shared__ float r[3 * FH];
    __shared__ float h[FH];
    __shared__ float lg[NC];
    int g = blockIdx.x, t = threadIdx.x;
    for (int i = t; i < 3 * FH; i += 128) r[i] = rep[g * 3 * FH + i];
    __syncthreads();
    float acc = b1[t];
    for (int k = 0; k < 3 * FH; ++k) acc += r[k] * W1[k * FH + t];
    h[t] = acc > 0.f ? acc : 0.f;
    __syncthreads();
    if (t < NC) {
        float a = b2[t];
        for (int k = 0; k < FH; ++k) a += h[k] * W2[k * NC + t];
        lg[t] = a;
    }
    __syncthreads();
    if (t == 0) {
        float mx = lg[0];
        for (int i = 1; i < NC; ++i) mx = fmaxf(mx, lg[i]);
        float se = 0.f;
        for (int i = 0; i < NC; ++i) se += expf(lg[i] - mx);
        float lse = mx + logf(se);
        for (int i = 0; i < NC; ++i) out[g * NC + i] = lg[i] - lse;
    }
}

// =====================================================================
extern "C" void kernel_launch(void* const* d_in, const int* in_sizes, int n_in,
                              void* d_out, int out_size, void* d_ws, size_t ws_size,
                              hipStream_t stream) {
    const float* x        = (const float*)d_in[0];
    const int*   ei       = (const int*)d_in[1];
    const int*   batch    = (const int*)d_in[2];
    const float* bn_gamma = (const float*)d_in[3];
    const float* bn_beta  = (const float*)d_in[4];
    const float* gcn_w    = (const float*)d_in[5];
    const float* gcn_b    = (const float*)d_in[6];
    const float* w_att    = (const float*)d_in[7];
    const float* lin1_w   = (const float*)d_in[8];
    const float* lin1_b   = (const float*)d_in[9];
    const float* lin2_w   = (const float*)d_in[10];
    const float* lin2_b   = (const float*)d_in[11];
    float* out = (float*)d_out;

    char* ws = (char*)d_ws;
    auto alloc = [&](size_t nf) {
        float* p = (float*)ws;
        ws += ((nf * sizeof(float) + 255) / 256) * 256;
        return p;
    };
    float* xbuf   = alloc((size_t)NN * FH);
    float* hbuf   = alloc((size_t)NN * FH);
    float* aggbuf = alloc((size_t)NN * FH);
    float* dinv   = alloc(NN);
    float* sums   = alloc(FH);
    float* sumsq  = alloc(FH);
    float* counts = alloc(NG);
    float* wsum   = alloc(NG * FH);
    float* msum   = alloc(NG * FH);
    float* mmax   = alloc(NG * FH);
    float* rep    = alloc(NG * 3 * FH);

    // init accumulators
    setup_init<<<(NG * 3 * FH + 255) / 256, 256, 0, stream>>>(rep, wsum, msum, mmax,
                                                              dinv, counts, sums, sumsq);
    // BatchNorm
    bn_stats<<<512, 128, 0, stream>>>(x, sums, sumsq);
    bn_apply<<<(NN * FH + 255) / 256, 256, 0, stream>>>(x, sums, sumsq, bn_gamma, bn_beta, xbuf);
    // degree / counts
    deg_edges<<<(EE + 255) / 256, 256, 0, stream>>>(ei, dinv);
    deg_to_dinv<<<(NN + 255) / 256, 256, 0, stream>>>(dinv);
    count_nodes<<<(NN + 255) / 256, 256, 0, stream>>>(batch, counts);

    for (int l = 0; l < NL; ++l) {
        const float* Wl = gcn_w + (size_t)l * FH * FH;
        const float* bl = gcn_b + (size_t)l * FH;
        const float* al = w_att + (size_t)l * FH;
        gemm_node_wmma<<<(NN + 127) / 128, 256, 0, stream>>>(xbuf, Wl, hbuf);
        agg_init<<<(NN * FH + 255) / 256, 256, 0, stream>>>(hbuf, dinv, bl, aggbuf);
        edge_scatter<<<(EE + 7) / 8, 256, 0, stream>>>(ei, dinv, hbuf, aggbuf);
        squash_readout<<<NN, 128, 0, stream>>>(aggbuf, batch, al, xbuf, wsum, msum, mmax);
        layer_accum<<<(NG * FH + 255) / 256, 256, 0, stream>>>(rep, wsum, msum, mmax, counts);
    }

    mlp_head<<<NG, 128, 0, stream>>>(rep, lin1_w, lin1_b, lin2_w, lin2_b, out);
}